// KNNContextModel_6047313953107
// MI455X (gfx1250) — compile-verified
//
#include <hip/hip_runtime.h>
#include <hip/hip_bf16.h>

#define NPTS  32768
#define BATCH 4
#define MPTS  8192
#define KNN   8
#define CIN   8
#define HDIM  128
#define TDIM  34            // 34^3 dense voxel table per batch (shifted coords 0..33)
#define TSZ   (BATCH*TDIM*TDIM*TDIM)
#define EPSV  1e-5f

typedef float v2f __attribute__((ext_vector_type(2)));
typedef float v8f __attribute__((ext_vector_type(8)));

__device__ __forceinline__ v8f wmma_f32_4(v2f a, v2f b, v8f c) {
  // 8 args: (neg_a, A, neg_b, B, c_mod, C, reuse_a, reuse_b)
  return __builtin_amdgcn_wmma_f32_16x16x4_f32(false, a, false, b, (short)0, c,
                                               false, false);
}

// ---------------- kNN: per-batch brute force, top-8 by (d2, idx) ------------
__global__ void knn_kernel(const int* __restrict__ coords, int* __restrict__ nbr) {
  __shared__ int sx[256], sy[256], sz[256];
  const int tid = threadIdx.x;
  const int blk = blockIdx.x;        // 128 blocks, 32 per batch
  const int batch = blk >> 5;
  const int q = batch * MPTS + (blk & 31) * 256 + tid;
  const int qx = coords[q * 4 + 1], qy = coords[q * 4 + 2], qz = coords[q * 4 + 3];
  int bd[KNN], bi[KNN];
#pragma unroll
  for (int i = 0; i < KNN; ++i) { bd[i] = 0x7fffffff; bi[i] = 0x7fffffff; }
  for (int t = 0; t < MPTS; t += 256) {
    const int ci = batch * MPTS + t + tid;
    sx[tid] = coords[ci * 4 + 1];
    sy[tid] = coords[ci * 4 + 2];
    sz[tid] = coords[ci * 4 + 3];
    __syncthreads();
#pragma unroll 4
    for (int i = 0; i < 256; ++i) {
      const int dx = qx - sx[i], dy = qy - sy[i], dz = qz - sz[i];
      const int d2 = dx * dx + dy * dy + dz * dz;
      const int idx = batch * MPTS + t + i;
      if (d2 < bd[KNN - 1] || (d2 == bd[KNN - 1] && idx < bi[KNN - 1])) {
        int cd = d2, cx = idx;
#pragma unroll
        for (int p = 0; p < KNN; ++p) {   // swap-insert, keeps list sorted asc
          const bool sw = (cd < bd[p]) || (cd == bd[p] && cx < bi[p]);
          if (sw) { int td = bd[p], ti = bi[p]; bd[p] = cd; bi[p] = cx; cd = td; cx = ti; }
        }
      }
    }
    __syncthreads();
  }
#pragma unroll
  for (int k = 0; k < KNN; ++k) nbr[q * KNN + k] = bi[k];
}

// ---------------- dense voxel hash table + 27-offset neighbor ids -----------
__global__ void table_init_kernel(int* __restrict__ table) {
  const int i = blockIdx.x * 256 + threadIdx.x;
  if (i < TSZ) table[i] = -1;
}

__global__ void table_scatter_kernel(const int* __restrict__ coords,
                                     int* __restrict__ table) {
  const int n = blockIdx.x * 256 + threadIdx.x;
  const int b = coords[n * 4 + 0];
  const int x = coords[n * 4 + 1] + 1;
  const int y = coords[n * 4 + 2] + 1;
  const int z = coords[n * 4 + 3] + 1;
  table[((b * TDIM + x) * TDIM + y) * TDIM + z] = n;
}

__global__ void nbh_kernel(const int* __restrict__ coords,
                           const int* __restrict__ table,
                           int* __restrict__ nbh) {
  const int n = blockIdx.x * 256 + threadIdx.x;
  const int b = coords[n * 4 + 0];
  const int x = coords[n * 4 + 1] + 1;
  const int y = coords[n * 4 + 2] + 1;
  const int z = coords[n * 4 + 3] + 1;
#pragma unroll
  for (int o = 0; o < 27; ++o) {      // meshgrid(ij) order: dx slowest (o/9), dz fastest
    const int dx = o / 9 - 1, dy = (o / 3) % 3 - 1, dz = o % 3 - 1;
    nbh[o * NPTS + n] = table[((b * TDIM + (x + dx)) * TDIM + (y + dy)) * TDIM + (z + dz)];
  }
}

// ---------------- GEMM1: combined[N,72] @ w1[72,128] via fp32 WMMA ----------
__global__ void gemm1_kernel(const float* __restrict__ feats,
                             const int* __restrict__ nbr,
                             const float* __restrict__ w1,
                             float* __restrict__ y1) {
  __shared__ float At[16 * 76];               // stride 76 avoids bank conflicts
  const int tid = threadIdx.x;
  const int rowBase = blockIdx.x * 16;
  for (int t = tid; t < 16 * 72; t += 256) {
    const int r = t / 72, e = t % 72;
    const int n = rowBase + r;
    float v;
    if (e < CIN) v = feats[n * CIN + e];
    else { const int k = (e - CIN) >> 3, c = (e - CIN) & 7; v = feats[nbr[n * KNN + k] * CIN + c]; }
    At[r * 76 + e] = v;
  }
  __syncthreads();
  const int wave = tid >> 5, lane = tid & 31;
  const int hi = (lane >= 16) ? 2 : 0;
  const int col = wave * 16 + (lane & 15);
  v8f acc = {0.f, 0.f, 0.f, 0.f, 0.f, 0.f, 0.f, 0.f};
#pragma unroll
  for (int k = 0; k < 18; ++k) {              // K = 72 = 18 * 4
    const int ac = 4 * k + hi;
    v2f a; a.x = At[(lane & 15) * 76 + ac]; a.y = At[(lane & 15) * 76 + ac + 1];
    v2f b; b.x = w1[ac * HDIM + col];        b.y = w1[(ac + 1) * HDIM + col];
    acc = wmma_f32_4(a, b, acc);
  }
#pragma unroll
  for (int r = 0; r < 8; ++r) {
    const int row = rowBase + r + ((lane >= 16) ? 8 : 0);
    y1[row * HDIM + col] = acc[r];
  }
}

// ---------------- per-channel sum / sumsq (for BN) --------------------------
__global__ void zero_stats_kernel(float* __restrict__ stats) {
  stats[threadIdx.x] = 0.f;                   // 512 threads zero 4*128 floats
}

__global__ void colstats_kernel(const float* __restrict__ y,
                                float* __restrict__ sum, float* __restrict__ ssq) {
  __shared__ float ssum[256], sss[256];
  const int tid = threadIdx.x;
  const int c = tid & 127, h = tid >> 7;
  const int base = blockIdx.x * 256;
  float s = 0.f, q = 0.f;
  for (int r = h; r < 256; r += 2) {
    const float v = y[(base + r) * HDIM + c];
    s += v; q += v * v;
  }
  ssum[tid] = s; sss[tid] = q;
  __syncthreads();
  if (tid < 128) {
    atomicAdd(&sum[c], ssum[tid] + ssum[tid + 128]);
    atomicAdd(&ssq[c], sss[tid] + sss[tid + 128]);
  }
}

// ---------------- conv3: sum_o gather(relu(bn(y1)))[o] @ w3[o] --------------
__global__ void conv3_kernel(const float* __restrict__ y1,
                             const float* __restrict__ sum1, const float* __restrict__ ssq1,
                             const float* __restrict__ g1,   const float* __restrict__ b1,
                             const int* __restrict__ nbh,
                             const float* __restrict__ w3,
                             float* __restrict__ y2) {
  __shared__ float At[64 * 132];              // 64 rows, stride 132
  __shared__ float sc[HDIM], sh[HDIM];
  const int tid = threadIdx.x;
  if (tid < HDIM) {
    const float mu = sum1[tid] * (1.f / NPTS);
    const float var = ssq1[tid] * (1.f / NPTS) - mu * mu;
    const float s = rsqrtf(var + EPSV) * g1[tid];
    sc[tid] = s; sh[tid] = b1[tid] - mu * s;
  }
  __syncthreads();
  const int rowBase = blockIdx.x * 64;
  const int wave = tid >> 5, lane = tid & 31;
  const int hi = (lane >= 16) ? 2 : 0;
  const int col = wave * 16 + (lane & 15);
  v8f acc[4];
#pragma unroll
  for (int rt = 0; rt < 4; ++rt) acc[rt] = (v8f){0.f,0.f,0.f,0.f,0.f,0.f,0.f,0.f};
  const int fr = tid >> 2;                    // fill row 0..63 (4 threads/row)
  const int fq = (tid & 3) * 32;              // fill col start
  for (int o = 0; o < 27; ++o) {
    const int j = nbh[o * NPTS + rowBase + fr];
    if (j >= 0) {
      const float* src = y1 + (size_t)j * HDIM + fq;
#pragma unroll
      for (int c = 0; c < 32; ++c) {
        const float hv = src[c] * sc[fq + c] + sh[fq + c];
        At[fr * 132 + fq + c] = hv > 0.f ? hv : 0.f;
      }
    } else {
#pragma unroll
      for (int c = 0; c < 32; ++c) At[fr * 132 + fq + c] = 0.f;
    }
    __syncthreads();
    const float* w3o = w3 + (size_t)o * HDIM * HDIM;
#pragma unroll 4
    for (int k = 0; k < 32; ++k) {            // hoist B across 4 row-tiles
      const int ac = 4 * k + hi;
      v2f b; b.x = w3o[ac * HDIM + col]; b.y = w3o[(ac + 1) * HDIM + col];
#pragma unroll
      for (int rt = 0; rt < 4; ++rt) {
        const int ar = rt * 16 + (lane & 15);
        v2f a; a.x = At[ar * 132 + ac]; a.y = At[ar * 132 + ac + 1];
        acc[rt] = wmma_f32_4(a, b, acc[rt]);
      }
    }
    __syncthreads();
  }
#pragma unroll
  for (int rt = 0; rt < 4; ++rt)
#pragma unroll
    for (int r = 0; r < 8; ++r) {
      const int row = rowBase + rt * 16 + r + ((lane >= 16) ? 8 : 0);
      y2[row * HDIM + col] = acc[rt][r];
    }
}

// ---------------- out: relu(bn(y2)) @ w_out[128,16] via WMMA ----------------
__global__ void out_kernel(const float* __restrict__ y2,
                           const float* __restrict__ sum2, const float* __restrict__ ssq2,
                           const float* __restrict__ g2,   const float* __restrict__ b2,
                           const float* __restrict__ wout,
                           float* __restrict__ out) {
  __shared__ float sc[HDIM], sh[HDIM];
  const int tid = threadIdx.x;
  if (tid < HDIM) {
    const float mu = sum2[tid] * (1.f / NPTS);
    const float var = ssq2[tid] * (1.f / NPTS) - mu * mu;
    const float s = rsqrtf(var + EPSV) * g2[tid];
    sc[tid] = s; sh[tid] = b2[tid] - mu * s;
  }
  __syncthreads();
  const int wave = tid >> 5, lane = tid & 31;
  const int hi = (lane >= 16) ? 2 : 0;
  const int tileBase = blockIdx.x * 128 + wave * 16;
  const int n = tileBase + (lane & 15);
  const int colo = lane & 15;                 // output col (N = 16 exactly)
  v8f acc = {0.f, 0.f, 0.f, 0.f, 0.f, 0.f, 0.f, 0.f};
#pragma unroll
  for (int k = 0; k < 32; ++k) {
    const int c = 4 * k + hi;
    const float v0 = y2[n * HDIM + c]     * sc[c]     + sh[c];
    const float v1 = y2[n * HDIM + c + 1] * sc[c + 1] + sh[c + 1];
    v2f a; a.x = v0 > 0.f ? v0 : 0.f; a.y = v1 > 0.f ? v1 : 0.f;
    v2f b; b.x = wout[c * 16 + colo]; b.y = wout[(c + 1) * 16 + colo];
    acc = wmma_f32_4(a, b, acc);
  }
#pragma unroll
  for (int r = 0; r < 8; ++r) {
    const int row = tileBase + r + ((lane >= 16) ? 8 : 0);
    out[row * 16 + colo] = acc[r];
  }
}

// ---------------- launch ----------------------------------------------------
extern "C" void kernel_launch(void* const* d_in, const int* in_sizes, int n_in,
                              void* d_out, int out_size, void* d_ws, size_t ws_size,
                              hipStream_t stream) {
  (void)in_sizes; (void)n_in; (void)out_size; (void)ws_size;
  const int*   coords = (const int*)  d_in[0];
  const float* feats  = (const float*)d_in[1];
  const float* w1     = (const float*)d_in[2];
  const float* g1     = (const float*)d_in[3];
  const float* b1     = (const float*)d_in[4];
  const float* w3     = (const float*)d_in[5];
  const float* g2     = (const float*)d_in[6];
  const float* b2     = (const float*)d_in[7];
  const float* wout   = (const float*)d_in[8];
  float* out = (float*)d_out;

  char* ws = (char*)d_ws;
  int*   nbr   = (int*)(ws + 0);             //  1,048,576 B
  int*   table = (int*)(ws + 1048576);       //    628,864 B
  int*   nbh   = (int*)(ws + 1677568);       //  3,538,944 B
  float* stats = (float*)(ws + 5216512);     //      2,048 B
  float* sum1 = stats, *ssq1 = stats + 128, *sum2 = stats + 256, *ssq2 = stats + 384;
  float* bufA  = (float*)(ws + 5218560);     // 16 MB  (y1)
  float* bufB  = bufA + (size_t)NPTS * HDIM; // 16 MB  (y2)

  knn_kernel<<<128, 256, 0, stream>>>(coords, nbr);
  table_init_kernel<<<(TSZ + 255) / 256, 256, 0, stream>>>(table);
  table_scatter_kernel<<<NPTS / 256, 256, 0, stream>>>(coords, table);
  nbh_kernel<<<NPTS / 256, 256, 0, stream>>>(coords, table, nbh);
  gemm1_kernel<<<NPTS / 16, 256, 0, stream>>>(feats, nbr, w1, bufA);
  zero_stats_kernel<<<1, 512, 0, stream>>>(stats);
  colstats_kernel<<<NPTS / 256, 256, 0, stream>>>(bufA, sum1, ssq1);
  conv3_kernel<<<NPTS / 64, 256, 0, stream>>>(bufA, sum1, ssq1, g1, b1, nbh, w3, bufB);
  colstats_kernel<<<NPTS / 256, 256, 0, stream>>>(bufB, sum2, ssq2);
  out_kernel<<<NPTS / 128, 256, 0, stream>>>(bufB, sum2, ssq2, g2, b2, wout, out);
}